// MultiHeadAttentionSequential_19756849561621
// MI455X (gfx1250) — compile-verified
//
#include <hip/hip_runtime.h>

typedef __attribute__((ext_vector_type(2))) float v2f;
typedef __attribute__((ext_vector_type(8))) float v8f;

#define HN 32
#define DD 4096
#define KK 128
#define VV 128
#define MM 8191
#define MC 8192  // M+1 rows after append (also logits length per head)

// workspace layout (float offsets)
#define WS_Q      0
#define WS_KNEW   (WS_Q + HN * KK)        // 4096
#define WS_VNEW   (WS_KNEW + HN * KK)     // 8192
#define WS_LOGITS (WS_VNEW + HN * VV)     // 12288
#define WS_O      (WS_LOGITS + HN * MC)   // 274432

// ---------------------------------------------------------------------------
// Generic batched matvec via V_WMMA_F32_16X16X4_F32.
//   y[h, n] = sum_d x[d] * W[h, d, n]   (W row-major Dred x N per batch)
// Vector x lives in row 0 of the 16x4 A tile (lanes 0/16 hold K pairs, all
// other lanes zero via branch-free cndmask). Row 0 of the accumulator
// (VGPR0, lanes 0-15) is the result. Grid exactly covers N: EXEC is all-1s
// at every WMMA.
// ---------------------------------------------------------------------------
__global__ void __launch_bounds__(128)
matvec16_wmma(const float* __restrict__ x, const float* __restrict__ W,
              float* __restrict__ y, int Dred, int N,
              long wStride, int yStride) {
  const int h = blockIdx.y;
  const int lane = threadIdx.x & 31;
  const int wv = threadIdx.x >> 5;
  const int n0 = (blockIdx.x * 4 + wv) * 16;
  const int nl = lane & 15;
  const int kb = (lane < 16) ? 0 : 2;   // K pair held by this half-wave
  const bool isA = (nl == 0);           // lanes 0 and 16 carry the A row
  const float* __restrict__ xp = x + kb;
  const float* __restrict__ bp = W + (long)h * wStride + (long)kb * N + n0 + nl;
  const long stepW = 4L * (long)N;
  v8f c = {};
  for (int d = 0; d < Dred; d += 4) {
    // branch-free A: every lane loads 8B (valid addr), non-carrier lanes -> 0
    float2 xv = *(const float2*)(xp + d);
    v2f a; a.x = isA ? xv.x : 0.f; a.y = isA ? xv.y : 0.f;
    v2f b; b.x = bp[0]; b.y = bp[N];            // B[k][n] = W[d+k][n0+n]
    __builtin_prefetch(bp + 2 * stepW, 0, 0);   // stream next k-slab into GL2
    c = __builtin_amdgcn_wmma_f32_16x16x4_f32(false, a, false, b, (short)0, c,
                                              false, false);
    bp += stepW;
  }
  if (lane < 16) y[h * yStride + n0 + nl] = c[0];  // row 0 of D
}

// ---------------------------------------------------------------------------
// Append k_new/v_new as cache row M and compute logits[h, M] = q . k_new
// ---------------------------------------------------------------------------
__global__ void __launch_bounds__(128)
append_kv(const float* __restrict__ q, const float* __restrict__ kn,
          const float* __restrict__ vn, float* __restrict__ Kc,
          float* __restrict__ Vc, float* __restrict__ logits) {
  const int h = blockIdx.x;
  const int t = threadIdx.x;  // 128 threads
  const float kval = kn[h * KK + t];
  Kc[(size_t)h * (MC * KK) + (MM * KK) + t] = kval;
  Vc[(size_t)h * (MC * VV) + (MM * VV) + t] = vn[h * VV + t];
  __shared__ float red[128];
  red[t] = q[h * KK + t] * kval;
  __syncthreads();
  for (int s = 64; s > 0; s >>= 1) {
    if (t < s) red[t] += red[t + s];
    __syncthreads();
  }
  if (t == 0) logits[h * MC + MM] = red[0];
}

// ---------------------------------------------------------------------------
// logits[h, m] = q[h] . prev_K[h, m]  (WMMA, 16 rows per wave) fused with the
// prev_K -> Kc copy (every loaded float2 is stored exactly once, NT store).
// ---------------------------------------------------------------------------
__global__ void __launch_bounds__(256)
attn_logits_copy(const float* __restrict__ q, const float* __restrict__ prevK,
                 float* __restrict__ Kc, float* __restrict__ logits) {
  const int h = blockIdx.y;
  const int lane = threadIdx.x & 31;
  const int wv = threadIdx.x >> 5;
  const int m0 = (blockIdx.x * 8 + wv) * 16;  // 64 blocks x 8 waves -> 8192 rows
  const int nl = lane & 15;
  const int kb = (lane < 16) ? 0 : 2;
  const int row = m0 + nl;
  const bool valid = row < MM;
  const int rowc = valid ? row : (MM - 1);
  const bool isA = (nl == 0);
  const float* __restrict__ qh = q + h * KK + kb;
  const float* __restrict__ Kh = prevK + (size_t)h * (MM * KK);
  float* __restrict__ Kch = Kc + (size_t)h * (MC * KK);
  const int ldo = (rowc << 7) + kb;  // 32-bit in-head load offset
  const int sto = (row << 7) + kb;   // 32-bit in-head store offset
  v8f c = {};
#pragma unroll 4
  for (int d = 0; d < KK; d += 4) {
    float2 qv = *(const float2*)(qh + d);
    v2f a; a.x = isA ? qv.x : 0.f; a.y = isA ? qv.y : 0.f;
    // B = K^T tile: B[k][n] = prev_K[m0+n][d+k]  (contiguous float2 per lane)
    float2 kv = *(const float2*)(Kh + ldo + d);
    v2f b; b.x = kv.x; b.y = kv.y;
    if (valid) {  // fused cache copy, write-once stream -> non-temporal
      v2f st; st.x = kv.x; st.y = kv.y;
      __builtin_nontemporal_store(st, (v2f*)(Kch + sto + d));
    }
    c = __builtin_amdgcn_wmma_f32_16x16x4_f32(false, a, false, b, (short)0, c,
                                              false, false);
  }
  if (lane < 16 && valid) logits[h * MC + row] = c[0];
}

// ---------------------------------------------------------------------------
// Softmax over MC logits per head (in place -> weights)
// ---------------------------------------------------------------------------
__global__ void __launch_bounds__(256)
softmax_kernel(float* __restrict__ logits) {
  const int h = blockIdx.x;
  float* L = logits + h * MC;
  const int t = threadIdx.x;
  __shared__ float red[256];
  float mx = -3.402823466e38f;
  for (int i = t; i < MC; i += 256) mx = fmaxf(mx, L[i]);
  red[t] = mx;
  __syncthreads();
  for (int s = 128; s > 0; s >>= 1) {
    if (t < s) red[t] = fmaxf(red[t], red[t + s]);
    __syncthreads();
  }
  const float m = red[0];
  __syncthreads();
  float sum = 0.f;
  for (int i = t; i < MC; i += 256) {
    const float e = __expf(L[i] - m);
    L[i] = e;
    sum += e;
  }
  red[t] = sum;
  __syncthreads();
  for (int s = 128; s > 0; s >>= 1) {
    if (t < s) red[t] += red[t + s];
    __syncthreads();
  }
  const float inv = 1.0f / red[0];
  for (int i = t; i < MC; i += 256) L[i] *= inv;
}

// ---------------------------------------------------------------------------
// o[h, v] += sum_m w[h, m] * prev_V[h, m, v]  (WMMA, 512-row partial chunks,
// atomicAdd into o) fused with the prev_V -> Vc copy (NT stores).
// Row MM's contribution (v_new) is added by add_new_v afterwards.
// ---------------------------------------------------------------------------
__global__ void __launch_bounds__(256)
attn_out_copy(const float* __restrict__ w, const float* __restrict__ prevV,
              float* __restrict__ Vc, float* __restrict__ o) {
  const int h = blockIdx.y;
  const int lane = threadIdx.x & 31;
  const int wv = threadIdx.x >> 5;  // 8 waves = 8 v-tiles of 16
  const int v0 = wv * 16;
  const int mbeg = blockIdx.x * 512;  // 16 chunks cover rows 0..8191
  const int nl = lane & 15;
  const int kb = (lane < 16) ? 0 : 2;
  const bool isA = (nl == 0);
  const int col = v0 + nl;
  const float* __restrict__ wh = w + h * MC + kb;
  const float* __restrict__ Vh = prevV + (size_t)h * (MM * VV);
  float* __restrict__ Vch = Vc + (size_t)h * (MC * VV);
  v8f c = {};
  for (int m = mbeg; m < mbeg + 512; m += 4) {
    const int r0 = m + kb;      // row of b.x: max 8190, always valid
    const int r1 = r0 + 1;      // row of b.y: 8191 only in last iteration
    float2 wv2 = *(const float2*)(wh + m);  // w[m+kb], w[m+kb+1] (idx<=8191 ok)
    v2f a;
    a.x = isA ? wv2.x : 0.f;
    a.y = (isA && r1 < MM) ? wv2.y : 0.f;   // row 8191 handled by add_new_v
    const int r1c = (r1 < MM) ? r1 : (MM - 1);
    v2f b;
    b.x = Vh[(r0 << 7) + col];              // B[k][n] = prev_V[m+k][v0+n]
    b.y = Vh[(r1c << 7) + col];
    __builtin_nontemporal_store(b.x, Vch + (r0 << 7) + col);  // fused copy
    if (r1 < MM) __builtin_nontemporal_store(b.y, Vch + (r1 << 7) + col);
    c = __builtin_amdgcn_wmma_f32_16x16x4_f32(false, a, false, b, (short)0, c,
                                              false, false);
  }
  if (lane < 16) atomicAdd(&o[h * VV + col], c[0]);
}

__global__ void __launch_bounds__(256)
add_new_v(const float* __restrict__ w, const float* __restrict__ vn,
          float* __restrict__ o) {
  const int i = blockIdx.x * blockDim.x + threadIdx.x;  // HN*VV = 4096
  const int h = i >> 7;
  o[i] += w[h * MC + MM] * vn[i];
}

// ---------------------------------------------------------------------------
extern "C" void kernel_launch(void* const* d_in, const int* in_sizes, int n_in,
                              void* d_out, int out_size, void* d_ws,
                              size_t ws_size, hipStream_t stream) {
  (void)in_sizes; (void)n_in; (void)out_size; (void)ws_size;
  const float* x     = (const float*)d_in[0];
  const float* prevK = (const float*)d_in[1];
  const float* prevV = (const float*)d_in[2];
  const float* Wq    = (const float*)d_in[3];
  const float* Wk    = (const float*)d_in[4];
  const float* Wv    = (const float*)d_in[5];
  const float* Wo    = (const float*)d_in[6];

  float* y  = (float*)d_out;                 // [D]
  float* Kc = y + DD;                        // [H, MC, K]
  float* Vc = Kc + (size_t)HN * MC * KK;     // [H, MC, V]

  float* ws     = (float*)d_ws;
  float* q      = ws + WS_Q;
  float* knew   = ws + WS_KNEW;
  float* vnew   = ws + WS_VNEW;
  float* logits = ws + WS_LOGITS;
  float* o      = ws + WS_O;

  // zero the o accumulator (atomicAdd targets)
  hipMemsetAsync(o, 0, (size_t)HN * VV * sizeof(float), stream);

  // QKV projections: y[h,n] = sum_d x[d] * W[h,d,n]
  dim3 bProj(128), gProj(KK / 16 / 4, HN);  // 2 x 32 blocks, 4 waves each
  matvec16_wmma<<<gProj, bProj, 0, stream>>>(x, Wq, q,    DD, KK, (long)DD * KK, KK);
  matvec16_wmma<<<gProj, bProj, 0, stream>>>(x, Wk, knew, DD, KK, (long)DD * KK, KK);
  matvec16_wmma<<<gProj, bProj, 0, stream>>>(x, Wv, vnew, DD, VV, (long)DD * VV, VV);

  // cache row 8191 + logits[8191]
  append_kv<<<HN, 128, 0, stream>>>(q, knew, vnew, Kc, Vc, logits);

  // logits for rows 0..8190, fused prev_K -> Kc copy
  attn_logits_copy<<<dim3(64, HN), 256, 0, stream>>>(q, prevK, Kc, logits);

  // softmax -> weights (in place)
  softmax_kernel<<<HN, 256, 0, stream>>>(logits);

  // o = weights . V (rows 0..8190), fused prev_V -> Vc copy
  attn_out_copy<<<dim3(16, HN), 256, 0, stream>>>(logits, prevV, Vc, o);
  add_new_v<<<16, 256, 0, stream>>>(logits, vnew, o);

  // y[d] = sum_{h,v} o[h,v] * Wo[h,v,d]  == (1x4096) @ (4096x4096)
  matvec16_wmma<<<dim3(DD / 16 / 4, 1), 128, 0, stream>>>(
      o, Wo, y, HN * VV, DD, 0L, 0);
}